// BilateralFilter_6622839570946
// MI455X (gfx1250) — compile-verified
//
#include <hip/hip_runtime.h>
#include <stdint.h>

typedef uint32_t u32x4 __attribute__((ext_vector_type(4)));
typedef int32_t  i32x4 __attribute__((ext_vector_type(4)));
typedef int32_t  i32x8 __attribute__((ext_vector_type(8)));

#define TS   32          // output tile side
#define RAD  3           // spatial radius
#define IMH  1024
#define IMW  1024
#define MAXT (TS + 2*RAD) // 38

// -50 * log2(e)  (color term folded into exp2)
#define CC (-72.13475204444817f)

__global__ __launch_bounds__(256) void
bilateral7x7_tdm_kernel(const float* __restrict__ in, float* __restrict__ out)
{
    __shared__ float tile[MAXT * MAXT];

    const int bx    = blockIdx.x;
    const int by    = blockIdx.y;
    const int plane = blockIdx.z;           // batch*3 + channel

    // Clamped halo window actually resident in memory (all reads in-bounds).
    const int ox0 = max(bx * TS - RAD, 0);
    const int ox1 = min(bx * TS + TS + RAD, IMW);
    const int oy0 = max(by * TS - RAD, 0);
    const int oy1 = min(by * TS + TS + RAD, IMH);
    const int lw  = ox1 - ox0;              // 35..38
    const int lh  = oy1 - oy0;              // 35..38

    const int tid = (int)threadIdx.x;

    // ---- Stage tile into LDS with the Tensor Data Mover (wave 0 only) ----
    if (__builtin_amdgcn_readfirstlane(tid) == 0) {
        const uint64_t gaddr = (uint64_t)(uintptr_t)in +
            (((uint64_t)plane * (uint64_t)(IMH * IMW)) +
             (uint64_t)oy0 * IMW + (uint64_t)ox0) * 4ull;
        const uint32_t ldsa = (uint32_t)(uintptr_t)&tile[0];

        // D# group 0: count=1 (valid), lds_addr, global_addr[56:0], type=2
        u32x4 g0;
        g0[0] = 1u;
        g0[1] = ldsa;
        g0[2] = (uint32_t)gaddr;
        g0[3] = ((uint32_t)(gaddr >> 32) & 0x01FFFFFFu) | 0x80000000u;

        // D# group 1: data_size=4B, tensor_dim0=lw, tensor_dim1=lh,
        //             tile_dim0=lw, tile_dim1=lh, dim0 stride = IMW elements
        i32x8 g1;
        g1[0] = 0x00020000;                                   // data_size=2 (4B)
        g1[1] = (int)(((uint32_t)lw & 0xFFFFu) << 16);        // tensor_dim0[15:0]
        g1[2] = (int)((((uint32_t)lw >> 16) & 0xFFFFu) |
                      (((uint32_t)lh & 0xFFFFu) << 16));      // td0 hi | td1 lo
        g1[3] = (int)((((uint32_t)lh >> 16) & 0xFFFFu) |
                      (((uint32_t)lw & 0xFFFFu) << 16));      // td1 hi | tile_dim0
        g1[4] = (int)((uint32_t)lh & 0xFFFFu);                // tile_dim1 (tile_dim2=0)
        g1[5] = IMW;                                          // tensor_dim0_stride lo
        g1[6] = 0;
        g1[7] = 0;

        const i32x4 gz4 = {0, 0, 0, 0};                       // 2D: groups 2/3 unused
        const i32x8 gz8 = {0, 0, 0, 0, 0, 0, 0, 0};
        __builtin_amdgcn_tensor_load_to_lds(g0, g1, gz4, gz4, gz8, 0);
        __builtin_amdgcn_s_wait_tensorcnt(0);
    }
    __syncthreads();

    // ---- Compute: 4 vertically adjacent pixels per thread ----
    const int tx  = tid & 31;               // column within tile
    const int tyg = tid >> 5;               // 0..7 row group
    const int gx  = bx * TS + tx;
    const int gy0 = by * TS + tyg * 4;

    // Clamped row offsets (in LDS) for raw rows gy0-3 .. gy0+6
    int rofs[10];
#pragma unroll
    for (int j = 0; j < 10; ++j) {
        int y = gy0 + j - 3;
        y = y < 0 ? 0 : (y > IMH - 1 ? IMH - 1 : y);
        rofs[j] = (y - oy0) * lw;
    }

    const int cxc = gx - ox0;               // center column offset (never clamps)
    const float c0 = tile[rofs[3] + cxc];
    const float c1 = tile[rofs[4] + cxc];
    const float c2 = tile[rofs[5] + cxc];
    const float c3 = tile[rofs[6] + cxc];

    float num0 = 0.f, num1 = 0.f, num2 = 0.f, num3 = 0.f;
    float den0 = 0.f, den1 = 0.f, den2 = 0.f, den3 = 0.f;

#pragma unroll
    for (int dx = 0; dx < 7; ++dx) {
        int x = gx + dx - 3;
        x = x < 0 ? 0 : (x > IMW - 1 ? IMW - 1 : x);
        const int cofs = x - ox0;

        float col[10];
#pragma unroll
        for (int j = 0; j < 10; ++j) col[j] = tile[rofs[j] + cofs];

#pragma unroll
        for (int dy = 0; dy < 7; ++dy) {
            // log2 of the spatial Gaussian weight: folds to an immediate
            const float sw = (float)(((dy - 3) * (dy - 3) + (dx - 3) * (dx - 3))
                                     * -0.08014972449383130);
#define TAP(i)                                                                 \
            {                                                                  \
                float s = col[dy + i];                                         \
                float d = s - c##i;                                            \
                float w = __builtin_amdgcn_exp2f(                              \
                              __builtin_fmaf(d * d, CC, sw));                  \
                num##i = __builtin_fmaf(w, s, num##i);                         \
                den##i += w;                                                   \
            }
            TAP(0) TAP(1) TAP(2) TAP(3)
#undef TAP
        }
    }

    const size_t base = (size_t)plane * (IMH * IMW) + (size_t)gy0 * IMW + gx;
    out[base + 0 * IMW] = num0 * __builtin_amdgcn_rcpf(den0);
    out[base + 1 * IMW] = num1 * __builtin_amdgcn_rcpf(den1);
    out[base + 2 * IMW] = num2 * __builtin_amdgcn_rcpf(den2);
    out[base + 3 * IMW] = num3 * __builtin_amdgcn_rcpf(den3);
}

extern "C" void kernel_launch(void* const* d_in, const int* in_sizes, int n_in,
                              void* d_out, int out_size, void* d_ws, size_t ws_size,
                              hipStream_t stream)
{
    (void)in_sizes; (void)n_in; (void)out_size; (void)d_ws; (void)ws_size;
    const float* in  = (const float*)d_in[0];
    float*       out = (float*)d_out;

    dim3 grid(IMW / TS, IMH / TS, 16 * 3);   // 32 x 32 tiles x 48 planes
    dim3 block(256, 1, 1);                   // 8 waves (wave32)
    bilateral7x7_tdm_kernel<<<grid, block, 0, stream>>>(in, out);
}